// RADSGCLayer_20117626814614
// MI455X (gfx1250) — compile-verified
//
#include <hip/hip_runtime.h>
#include <stdint.h>

#define D 128            // feature dim
#define WAVES_PER_BLOCK 8
#define EDGES_PER_WAVE 8
#define NBUF 4           // LDS row buffers per wave (power of 2)
#define PIPE 3           // async prefetch depth (NBUF - 1)

// ---------------------------------------------------------------------------
// deg[dst[e]] += 1  (f32 atomics; counts are exact in f32 far below 2^24)
// ---------------------------------------------------------------------------
__global__ __launch_bounds__(256) void sgc_degree(const int* __restrict__ dst,
                                                  float* __restrict__ deg, int E) {
  int e = blockIdx.x * blockDim.x + threadIdx.x;
  if (e < E) atomicAdd(&deg[dst[e]], 1.0f);
}

// norm[i] = rsqrt(max(deg[i], 1))  (in place)
__global__ __launch_bounds__(256) void sgc_norm(float* __restrict__ deg, int N) {
  int i = blockIdx.x * blockDim.x + threadIdx.x;
  if (i < N) deg[i] = rsqrtf(fmaxf(deg[i], 1.0f));
}

// ---------------------------------------------------------------------------
// out[dst[e], :] += h[src[e], :] * nrm[src[e]]
// One wave per 8 edges. Each 512B source row is staged into LDS with ONE
// wave-wide global_load_async_to_lds_b128 (32 lanes x 16B); a 4-buffer,
// depth-3 async pipeline (s_wait_asynccnt with static immediates) keeps 3
// random row-gathers in flight while the wave scatters with f32 atomics.
// Edge indices are loaded cooperatively (lanes 0-7: src, lanes 8-15: dst)
// and broadcast via __shfl. Edge index is wave-uniform -> EXEC all-ones at
// every async issue.
// ---------------------------------------------------------------------------
__global__ __launch_bounds__(256) void sgc_scatter(
    const float* __restrict__ h,    // [N, D] rows to gather
    const float* __restrict__ nrm,  // [N]
    const int* __restrict__ src,
    const int* __restrict__ dst,
    float* __restrict__ out,        // [N, D] zero-initialized accumulator
    int E) {
  __shared__ __align__(16) float smem[WAVES_PER_BLOCK * NBUF * D];  // 16 KB
  const int lane = threadIdx.x & 31;
  const int wave = threadIdx.x >> 5;
  float* wbuf = &smem[wave * NBUF * D];

  long long e0 = ((long long)blockIdx.x * WAVES_PER_BLOCK + wave) * EDGES_PER_WAVE;
  if (e0 >= E) return;  // uniform per wave
  int cnt = (int)(((long long)E - e0) < EDGES_PER_WAVE ? (E - e0) : EDGES_PER_WAVE);

  // Cooperative edge-list load: lanes 0..7 -> src, lanes 8..15 -> dst.
  int ev = 0;
  if (lane < 16) {
    int k = lane & 7;
    if (k < cnt) ev = (lane < 8) ? src[e0 + k] : dst[e0 + k];
  }
  auto Sn = [&](int i) { return __shfl(ev, i, 32); };
  auto Dn = [&](int i) { return __shfl(ev, 8 + i, 32); };

  auto stage = [&](int i) {
    int s = Sn(i);
    const float* g = h + (long long)s * D + lane * 4;
    unsigned lo = (unsigned)(uintptr_t)(&wbuf[(i & (NBUF - 1)) * D + lane * 4]);
    asm volatile("global_load_async_to_lds_b128 %0, %1, off"
                 :: "v"(lo), "v"((uint64_t)(uintptr_t)g) : "memory");
  };
  auto consume = [&](int i) {
    int s = Sn(i);
    int dn = Dn(i);
    float ns = nrm[s];
    float4 v = *(const float4*)&wbuf[(i & (NBUF - 1)) * D + lane * 4];
    float* o = out + (long long)dn * D + lane * 4;
    atomicAdd(o + 0, v.x * ns);
    atomicAdd(o + 1, v.y * ns);
    atomicAdd(o + 2, v.z * ns);
    atomicAdd(o + 3, v.w * ns);
  };

  if (cnt == EDGES_PER_WAVE) {
    // Steady state (E % 8 == 0 for this problem): fully unrolled, depth-3
    // pipeline with compile-time wait immediates.
    stage(0); stage(1); stage(2);
#pragma unroll
    for (int i = 0; i < EDGES_PER_WAVE; ++i) {
      if (i + PIPE < EDGES_PER_WAVE) stage(i + PIPE);
      // Loads younger than load i that are still outstanding:
      int rem = EDGES_PER_WAVE - 1 - i;
      int wv = rem < PIPE ? rem : PIPE;
      if (wv == 3)      asm volatile("s_wait_asynccnt 0x3" ::: "memory");
      else if (wv == 2) asm volatile("s_wait_asynccnt 0x2" ::: "memory");
      else if (wv == 1) asm volatile("s_wait_asynccnt 0x1" ::: "memory");
      else              asm volatile("s_wait_asynccnt 0x0" ::: "memory");
      consume(i);
    }
  } else {
    // Ragged tail: simple stage->wait->consume.
    for (int i = 0; i < cnt; ++i) {
      stage(i);
      asm volatile("s_wait_asynccnt 0x0" ::: "memory");
      consume(i);
    }
  }
}

// ---------------------------------------------------------------------------
// After hop 1: h1 = A*norm ; out = feat + h1 ; A <- h1 (input to hop 2)
// One thread per float4.
// ---------------------------------------------------------------------------
__global__ __launch_bounds__(256) void sgc_combine1(
    const float* __restrict__ feat, const float* __restrict__ nrm,
    float* __restrict__ A, float* __restrict__ out, int N) {
  int t = blockIdx.x * blockDim.x + threadIdx.x;  // float4 index
  int total = N * (D / 4);
  if (t >= total) return;
  float w = nrm[t / (D / 4)];
  float4 a = ((const float4*)A)[t];
  float4 f = ((const float4*)feat)[t];
  float4 h1 = make_float4(a.x * w, a.y * w, a.z * w, a.w * w);
  ((float4*)out)[t] = make_float4(f.x + h1.x, f.y + h1.y, f.z + h1.z, f.w + h1.w);
  ((float4*)A)[t] = h1;
}

// After hop 2: out = (out + B*norm) / 3
__global__ __launch_bounds__(256) void sgc_combine2(
    const float* __restrict__ B, const float* __restrict__ nrm,
    float* __restrict__ out, int N) {
  int t = blockIdx.x * blockDim.x + threadIdx.x;
  int total = N * (D / 4);
  if (t >= total) return;
  float w = nrm[t / (D / 4)];
  const float inv3 = 1.0f / 3.0f;
  float4 bv = ((const float4*)B)[t];
  float4 o = ((float4*)out)[t];
  ((float4*)out)[t] = make_float4((o.x + bv.x * w) * inv3,
                                  (o.y + bv.y * w) * inv3,
                                  (o.z + bv.z * w) * inv3,
                                  (o.w + bv.w * w) * inv3);
}

// ---------------------------------------------------------------------------
extern "C" void kernel_launch(void* const* d_in, const int* in_sizes, int n_in,
                              void* d_out, int out_size, void* d_ws, size_t ws_size,
                              hipStream_t stream) {
  const float* feat = (const float*)d_in[0];
  const int* src = (const int*)d_in[1];
  const int* dst = (const int*)d_in[2];
  float* out = (float*)d_out;
  const int N = in_sizes[0] / D;   // 100000
  const int E = in_sizes[1];       // 640000

  // Workspace layout: [ norm: N f32 | A: N*D f32 | B: N*D f32 ]  (~103 MB)
  char* ws = (char*)d_ws;
  size_t nrm_bytes = (((size_t)N * sizeof(float)) + 255) & ~(size_t)255;
  float* nrm = (float*)ws;
  float* A = (float*)(ws + nrm_bytes);
  float* B = A + (size_t)N * D;
  size_t rowbytes = (size_t)N * D * sizeof(float);

  hipMemsetAsync(nrm, 0, (size_t)N * sizeof(float), stream);
  hipMemsetAsync(A, 0, rowbytes, stream);
  hipMemsetAsync(B, 0, rowbytes, stream);

  const int tb = 256;
  sgc_degree<<<(E + tb - 1) / tb, tb, 0, stream>>>(dst, nrm, E);
  sgc_norm<<<(N + tb - 1) / tb, tb, 0, stream>>>(nrm, N);

  long long waves = ((long long)E + EDGES_PER_WAVE - 1) / EDGES_PER_WAVE;
  int sblocks = (int)((waves + WAVES_PER_BLOCK - 1) / WAVES_PER_BLOCK);
  int total4 = N * (D / 4);
  int cblocks = (total4 + tb - 1) / tb;

  // Hop 1
  sgc_scatter<<<sblocks, tb, 0, stream>>>(feat, nrm, src, dst, A, E);
  sgc_combine1<<<cblocks, tb, 0, stream>>>(feat, nrm, A, out, N);

  // Hop 2
  sgc_scatter<<<sblocks, tb, 0, stream>>>(A, nrm, src, dst, B, E);
  sgc_combine2<<<cblocks, tb, 0, stream>>>(B, nrm, out, N);
}